// ValidationLayer_738734375311
// MI455X (gfx1250) — compile-verified
//
#include <hip/hip_runtime.h>
#include <hip/hip_bf16.h>
#include <stdint.h>

// Problem shape from the reference: B=64, S=512, H=1024 (float32).
#define BATCH 64
#define SEQ   512
#define HID   1024   // floats per row; 4096 bytes = 256 lanes * 16B

// ---------------------------------------------------------------------------
// Kernel 1: per batch row, stable compaction order via LDS prefix scan.
// order[b*SEQ + d] = source position s for destination slot d, or -1 (zero).
// ---------------------------------------------------------------------------
__global__ void build_order_kernel(const int* __restrict__ valid_ids,
                                   int* __restrict__ order) {
    const int b = blockIdx.x;          // one block per batch row
    const int t = threadIdx.x;         // 0..SEQ-1

    __shared__ int sc[SEQ];
    __shared__ int ord[SEQ];

    const int m = (valid_ids[b * SEQ + t] == 1) ? 1 : 0;
    sc[t]  = m;
    ord[t] = -1;
    __syncthreads();

    // Hillis-Steele inclusive scan over SEQ elements.
    #pragma unroll
    for (int off = 1; off < SEQ; off <<= 1) {
        int v = (t >= off) ? sc[t - off] : 0;
        __syncthreads();
        sc[t] += v;
        __syncthreads();
    }

    // Stable scatter: valid position t goes to destination (exclusive prefix).
    if (m) ord[sc[t] - 1] = t;
    __syncthreads();

    order[b * SEQ + t] = ord[t];
}

// ---------------------------------------------------------------------------
// Kernel 2: one block per output row. Copy a 4KB row through the CDNA5 async
// global->LDS->global DMA path (16B per lane), or write zeros for the tail.
// Each lane uses a private 16B LDS slice, so only an ASYNCcnt wait is needed
// between the async load and the async store (no block barrier).
// ---------------------------------------------------------------------------
__global__ __launch_bounds__(256)
void gather_rows_kernel(const float* __restrict__ src,
                        const int* __restrict__ order,
                        float* __restrict__ out) {
    const int row = blockIdx.x;        // row = b*SEQ + d
    const int b   = row >> 9;          // SEQ == 512
    const int t   = threadIdx.x;       // 0..255, one float4 each

    const int s = order[row];          // uniform across the block

    float4* dst4 = reinterpret_cast<float4*>(out + (size_t)row * HID);

    if (s >= 0) {
        const float4* src4 =
            reinterpret_cast<const float4*>(src + ((size_t)b * SEQ + (size_t)s) * HID);

        __shared__ __align__(16) unsigned char smem[256 * 16];

        // Per-lane LDS byte address (flat address truncated to LDS offset).
        const uint32_t lds  = (uint32_t)(uintptr_t)(smem + t * 16);
        const uint64_t gsrc = (uint64_t)(uintptr_t)(src4 + t);
        const uint64_t gdst = (uint64_t)(uintptr_t)(dst4 + t);

        // HBM -> LDS (async DMA path, 16B per lane)
        asm volatile("global_load_async_to_lds_b128 %0, %1, off"
                     :: "v"(lds), "v"(gsrc) : "memory");
        // Wait for the async load to land in LDS before sourcing the store.
        asm volatile("s_wait_asynccnt 0x0" ::: "memory");
        // LDS -> HBM (async DMA path, 16B per lane)
        asm volatile("global_store_async_from_lds_b128 %0, %1, off"
                     :: "v"(gdst), "v"(lds) : "memory");
        // Ensure the async store is retired before the wave can exit.
        asm volatile("s_wait_asynccnt 0x0" ::: "memory");
    } else {
        dst4[t] = make_float4(0.0f, 0.0f, 0.0f, 0.0f);
    }
}

// ---------------------------------------------------------------------------
// Launch wrapper.
//   d_in[0]: sequence_output  float32 [B, S, H]
//   d_in[1]: valid_ids        int32   [B, S]
//   d_out  : float32 [B, S, H]
//   d_ws   : >= B*S*4 bytes (gather order)
// ---------------------------------------------------------------------------
extern "C" void kernel_launch(void* const* d_in, const int* in_sizes, int n_in,
                              void* d_out, int out_size, void* d_ws, size_t ws_size,
                              hipStream_t stream) {
    const float* seq   = (const float*)d_in[0];
    const int*   valid = (const int*)d_in[1];
    float*       out   = (float*)d_out;
    int*         order = (int*)d_ws;   // BATCH*SEQ ints = 128 KiB

    (void)in_sizes; (void)n_in; (void)out_size; (void)ws_size;

    build_order_kernel<<<BATCH, SEQ, 0, stream>>>(valid, order);
    gather_rows_kernel<<<BATCH * SEQ, 256, 0, stream>>>(seq, order, out);
}